// l_P_58308476011188
// MI455X (gfx1250) — compile-verified
//
#include <hip/hip_runtime.h>
#include <stdint.h>

#define N_NODES 100000
#define D_FEAT  32

// ---------------------------------------------------------------------------
// Scatter-sum: one wave32 processes a chunk of 32 edges.
//  - Stage the chunk's row/col int64 pairs into LDS with the CDNA5 async
//    data mover (global_load_async_to_lds_b64, tracked by ASYNCcnt).
//  - Then loop the 32 edges: uniform LDS read of (r,c), coalesced 128B gather
//    of x[c][0..31] (L2-resident, 12.8MB), 32-lane f32 atomic add into
//    out[r][0..31] (L2-resident).
// ---------------------------------------------------------------------------
__global__ __launch_bounds__(256) void scatter_sum_kernel(
    const long long* __restrict__ row,
    const long long* __restrict__ col,
    const float*     __restrict__ x,
    float*           __restrict__ out,
    int n_edges)
{
    __shared__ long long lds_row[8 * 32];
    __shared__ long long lds_col[8 * 32];

    const int lane  = threadIdx.x & 31;
    const int wave  = threadIdx.x >> 5;
    const int gwave = blockIdx.x * 8 + wave;
    const int e0    = gwave * 32;
    if (e0 >= n_edges) return;

    long long* wrow = &lds_row[wave * 32];
    long long* wcol = &lds_col[wave * 32];

    if (e0 + 32 <= n_edges) {
        // ---- async stage: 32 x int64 row + 32 x int64 col -> LDS ----------
        const long long* gr = row + e0 + lane;
        const long long* gc = col + e0 + lane;
        unsigned ldsr = (unsigned)(uintptr_t)(void*)&wrow[lane];
        unsigned ldsc = (unsigned)(uintptr_t)(void*)&wcol[lane];
        asm volatile(
            "global_load_async_to_lds_b64 %0, %2, off\n\t"
            "global_load_async_to_lds_b64 %1, %3, off\n\t"
            "s_wait_asynccnt 0"
            :
            : "v"(ldsr), "v"(ldsc), "v"(gr), "v"(gc)
            : "memory");

        // Only the low dword of each int64 index is meaningful (< 100000).
        const int* wr = (const int*)wrow;
        const int* wc = (const int*)wcol;

        #pragma unroll 4
        for (int k = 0; k < 32; ++k) {
            const int r = wr[2 * k];           // uniform ds_load (broadcast)
            const int c = wc[2 * k];
            const float v = x[(size_t)c * D_FEAT + lane];   // coalesced 128B
            atomicAdd(&out[(size_t)r * D_FEAT + lane], v);  // L2 f32 atomic
        }
    } else {
        // tail chunk (not hit for E = 1.6M, kept for safety)
        for (int k = 0; k < 32; ++k) {
            if (e0 + k >= n_edges) break;
            const int r = (int)row[e0 + k];
            const int c = (int)col[e0 + k];
            const float v = x[(size_t)c * D_FEAT + lane];
            atomicAdd(&out[(size_t)r * D_FEAT + lane], v);
        }
    }
}

// ---------------------------------------------------------------------------
// Degree: one thread per edge; read only the low dword of the int64 row id.
// ---------------------------------------------------------------------------
__global__ __launch_bounds__(256) void degree_kernel(
    const long long* __restrict__ row,
    float*           __restrict__ deg,
    int n_edges)
{
    const int t = blockIdx.x * blockDim.x + threadIdx.x;
    if (t < n_edges) {
        const int r = ((const int*)row)[2 * t];
        atomicAdd(&deg[r], 1.0f);
    }
}

// ---------------------------------------------------------------------------
// Finalize: out[node][feat] /= max(deg[node], 1)
// ---------------------------------------------------------------------------
__global__ __launch_bounds__(256) void finalize_kernel(
    float*       __restrict__ out,
    const float* __restrict__ deg,
    int total)
{
    const int t = blockIdx.x * blockDim.x + threadIdx.x;
    if (t < total) {
        const float d = fmaxf(deg[t >> 5], 1.0f);
        out[t] = out[t] / d;
    }
}

extern "C" void kernel_launch(void* const* d_in, const int* in_sizes, int n_in,
                              void* d_out, int out_size, void* d_ws, size_t ws_size,
                              hipStream_t stream)
{
    const float*     x    = (const float*)d_in[0];
    const long long* edge = (const long long*)d_in[1];   // int64 [2, E] flat
    const int n_edges     = in_sizes[1] / 2;
    const long long* row  = edge;
    const long long* col  = edge + n_edges;

    float* out = (float*)d_out;       // [N_NODES, D_FEAT] f32
    float* deg = (float*)d_ws;        // [N_NODES] f32 scratch

    // Harness poisons d_out/d_ws; zero the accumulators every call.
    hipMemsetAsync(out, 0, (size_t)out_size * sizeof(float), stream);
    hipMemsetAsync(deg, 0, (size_t)N_NODES * sizeof(float), stream);

    const int waves  = (n_edges + 31) / 32;
    const int blocks = (waves + 7) / 8;          // 8 waves (256 thr) per block
    scatter_sum_kernel<<<blocks, 256, 0, stream>>>(row, col, x, out, n_edges);

    degree_kernel<<<(n_edges + 255) / 256, 256, 0, stream>>>(row, deg, n_edges);

    finalize_kernel<<<(out_size + 255) / 256, 256, 0, stream>>>(out, deg, out_size);
}